// ShuffleNetUnit_69879117906311
// MI455X (gfx1250) — compile-verified
//
#include <hip/hip_runtime.h>
#include <math.h>

// CDNA5 (gfx1250) wave32 WMMA types for V_WMMA_F32_16X16X4_F32:
//   A: 16x4 f32  -> 2 VGPRs/lane (v2f)
//   B: 4x16 f32  -> 2 VGPRs/lane (v2f)
//   C/D: 16x16 f32 -> 8 VGPRs/lane (v8f)
typedef float v2f __attribute__((ext_vector_type(2)));
typedef float v8f __attribute__((ext_vector_type(8)));

#define BN_EPS 1e-5f

// ---------------------------------------------------------------------------
// Kernel 1: grouped 1x1 conv (G=3, 160 in -> 40 out per group) + BN + ReLU.
// GEMM per (group, batch): D[40 x 784] = W[40 x 160] * X[160 x 784].
// One wave computes ALL 3 m-tiles (rows 0..47, 40 valid) for one 16-column
// tile: the B tile (x activations, the 96 MB tensor) is loaded ONCE per wave
// and shared across 3 accumulators. Steady state: 3 A-b64 + 2 B-b32 loads
// per 3 wmma. grid = (49 n-tiles, 3 groups, 64 batch), block = 32 (1 wave).
//
// WMMA f32 16x16x4 lane layout (ISA 7.12.2):
//   A: lanes 0-15 -> M=lane, {VGPR0,VGPR1} = K={0,1}; lanes 16-31 -> K={2,3}
//   B: VGPR0 = row K=(0|2), VGPR1 = row K=(1|3); N = lane&15
//   C/D: VGPR v -> row m0+v (lanes 0-15) / m0+v+8 (lanes 16-31), col = lane&15
// ---------------------------------------------------------------------------
__global__ __launch_bounds__(32) void k_conv1(
    const float* __restrict__ x,  const float* __restrict__ w1,
    const float* __restrict__ g1, const float* __restrict__ b1,
    const float* __restrict__ m1, const float* __restrict__ v1,
    float* __restrict__ t1)
{
    const int lane  = threadIdx.x;
    const int lo    = lane & 15;          // tile column / M within half
    const int hi    = (lane >> 4) & 1;    // lane half
    const int khalf = hi * 2;             // K offset of this lane half

    const int nt = blockIdx.x;            // 16-column tile of the 784 pixels
    const int g  = blockIdx.y;            // channel group
    const int n  = blockIdx.z;            // batch

    const int col = nt * 16 + lo;         // 0..783

    // A-matrix row pointers for the 3 m-tiles. Clamp rows 40..47 of the
    // partial tile so loads stay in-bounds; stores are guarded below.
    const float* wrow[3];
#pragma unroll
    for (int t = 0; t < 3; ++t) {
        int am = t * 16 + lo; if (am > 39) am = 39;
        wrow[t] = w1 + (size_t)(g * 40 + am) * 160;          // K-contiguous
    }
    const float* xg = x + ((size_t)n * 480 + (size_t)g * 160) * 784;

    v8f acc[3] = {};
#pragma unroll 2
    for (int k = 0; k < 160; k += 4) {
        // B: shared across the 3 accumulators (one k-pair, plane stride 784).
        v2f b;
        b.x = xg[(size_t)(k + khalf    ) * 784 + col];
        b.y = xg[(size_t)(k + khalf + 1) * 784 + col];
#pragma unroll
        for (int t = 0; t < 3; ++t) {
            v2f a = *(const v2f*)(wrow[t] + k + khalf);      // 8B-aligned
            acc[t] = __builtin_amdgcn_wmma_f32_16x16x4_f32(
                         false, a, false, b, (short)0, acc[t], false, false);
        }
    }

    // Epilogue: folded BN + ReLU. Validity is uniform per lane-half:
    // rows t*16 + v + hi*8 are all < 40 iff t*16 + hi*8 <= 32.
    float* out = t1 + (size_t)n * 120 * 784;
#pragma unroll
    for (int t = 0; t < 3; ++t) {
        if (t < 2 || hi == 0) {
#pragma unroll
            for (int v = 0; v < 8; ++v) {
                int gm   = g * 40 + t * 16 + v + hi * 8;     // global mid chan
                float s  = g1[gm] * rsqrtf(v1[gm] + BN_EPS);
                float bb = b1[gm] - m1[gm] * s;
                float val = acc[t][v] * s + bb;
                val = val > 0.f ? val : 0.f;
                out[(size_t)gm * 784 + col] = val;
            }
        }
    }
}

// ---------------------------------------------------------------------------
// Kernel 2: depthwise 3x3 (pad 1) + BN (no ReLU). 0.11 GFLOP, t1/t2 are
// L2-resident (24 MB each in 192 MB L2), so a simple VALU kernel suffices.
// ---------------------------------------------------------------------------
__global__ __launch_bounds__(256) void k_dw(
    const float* __restrict__ t1, const float* __restrict__ w2,
    const float* __restrict__ g2, const float* __restrict__ b2,
    const float* __restrict__ m2, const float* __restrict__ v2,
    float* __restrict__ t2)
{
    const int total = 64 * 120 * 784;
    int idx = blockIdx.x * 256 + threadIdx.x;
    if (idx >= total) return;

    int hw = idx % 784;
    int c  = (idx / 784) % 120;
    int n  = idx / (784 * 120);
    int h  = hw / 28, w = hw % 28;

    const float* in = t1 + ((size_t)n * 120 + c) * 784;
    const float* wt = w2 + c * 9;

    float sum = 0.f;
#pragma unroll
    for (int dh = -1; dh <= 1; ++dh) {
        int hh = h + dh;
        if ((unsigned)hh < 28u) {
#pragma unroll
            for (int dw = -1; dw <= 1; ++dw) {
                int ww = w + dw;
                if ((unsigned)ww < 28u)
                    sum += in[hh * 28 + ww] * wt[(dh + 1) * 3 + (dw + 1)];
            }
        }
    }
    float s = g2[c] * rsqrtf(v2[c] + BN_EPS);
    t2[idx] = sum * s + (b2[c] - m2[c] * s);
}

// ---------------------------------------------------------------------------
// Kernel 3: channel shuffle (folded into B-load indexing) + grouped 1x1 conv
// (40 in -> 160 out per group) + BN + ReLU + identity shortcut.
// GEMM per (group, batch): D[160 x 784] = W[160 x 40] * Xs[40 x 784].
// One wave computes 5 of the 10 m-tiles, sharing each B tile across 5
// accumulators (5 A-b64 + 2 B-b32 loads per 5 wmma).
// Shuffle fold: conv3 group g, within-group input j reads shuffled channel
// c = g*40 + j, whose pre-shuffle source is (c % 3)*40 + c/3.
// grid = (49 n-tiles, 3 groups * 2 m-halves, 64 batch), block = 32.
// ---------------------------------------------------------------------------
__global__ __launch_bounds__(32) void k_conv3(
    const float* __restrict__ t2, const float* __restrict__ w3,
    const float* __restrict__ g3, const float* __restrict__ b3,
    const float* __restrict__ m3, const float* __restrict__ v3,
    const float* __restrict__ x,  float* __restrict__ out)
{
    const int lane  = threadIdx.x;
    const int lo    = lane & 15;
    const int hi    = (lane >> 4) & 1;
    const int khalf = hi * 2;

    const int nt = blockIdx.x;
    const int g  = blockIdx.y / 2;        // channel group
    const int mh = blockIdx.y % 2;        // which 5 m-tiles (rows 0-79 / 80-159)
    const int n  = blockIdx.z;

    const int col = nt * 16 + lo;

    const float* wrow[5];
#pragma unroll
    for (int t = 0; t < 5; ++t) {
        int m0 = (mh * 5 + t) * 16;                           // within group
        wrow[t] = w3 + (size_t)(g * 160 + m0 + lo) * 40;      // K-contiguous
    }
    const float* tin = t2 + (size_t)n * 120 * 784;

    v8f acc[5] = {};
#pragma unroll 2
    for (int k = 0; k < 40; k += 4) {
        // Shuffled channel indices for this K pair (shared across 5 tiles).
        int c0 = g * 40 + k + khalf;
        int c1 = c0 + 1;
        int s0 = (c0 % 3) * 40 + c0 / 3;
        int s1 = (c1 % 3) * 40 + c1 / 3;
        v2f b;
        b.x = tin[(size_t)s0 * 784 + col];
        b.y = tin[(size_t)s1 * 784 + col];
#pragma unroll
        for (int t = 0; t < 5; ++t) {
            v2f a = *(const v2f*)(wrow[t] + k + khalf);
            acc[t] = __builtin_amdgcn_wmma_f32_16x16x4_f32(
                         false, a, false, b, (short)0, acc[t], false, false);
        }
    }

    const float* xin = x   + (size_t)n * 480 * 784;
    float*       o   = out + (size_t)n * 480 * 784;
#pragma unroll
    for (int t = 0; t < 5; ++t) {
#pragma unroll
        for (int v = 0; v < 8; ++v) {
            int m    = g * 160 + (mh * 5 + t) * 16 + v + hi * 8;  // < 480
            float s  = g3[m] * rsqrtf(v3[m] + BN_EPS);
            float bb = b3[m] - m3[m] * s;
            float val = acc[t][v] * s + bb;
            val = val > 0.f ? val : 0.f;
            o[(size_t)m * 784 + col] = val + xin[(size_t)m * 784 + col];
        }
    }
}

// ---------------------------------------------------------------------------
// Launcher. Workspace layout: t1 (24 MB) | t2 (24 MB)  -> needs ~48.2 MB.
// ---------------------------------------------------------------------------
extern "C" void kernel_launch(void* const* d_in, const int* in_sizes, int n_in,
                              void* d_out, int out_size, void* d_ws, size_t ws_size,
                              hipStream_t stream)
{
    const float* x  = (const float*)d_in[0];
    const float* w1 = (const float*)d_in[1];
    const float* g1 = (const float*)d_in[2];
    const float* b1 = (const float*)d_in[3];
    const float* m1 = (const float*)d_in[4];
    const float* v1 = (const float*)d_in[5];
    const float* w2 = (const float*)d_in[6];
    const float* g2 = (const float*)d_in[7];
    const float* b2 = (const float*)d_in[8];
    const float* m2 = (const float*)d_in[9];
    const float* v2 = (const float*)d_in[10];
    const float* w3 = (const float*)d_in[11];
    const float* g3 = (const float*)d_in[12];
    const float* b3 = (const float*)d_in[13];
    const float* m3 = (const float*)d_in[14];
    const float* v3 = (const float*)d_in[15];

    float* t1 = (float*)d_ws;
    float* t2 = t1 + (size_t)64 * 120 * 784;

    // Stage 1: grouped 1x1 GEMM (WMMA f32, 3 acc tiles/wave) + BN + ReLU
    k_conv1<<<dim3(49, 3, 64), dim3(32), 0, stream>>>(x, w1, g1, b1, m1, v1, t1);

    // Stage 2: depthwise 3x3 + BN
    const int total = 64 * 120 * 784;
    k_dw<<<dim3((total + 255) / 256), dim3(256), 0, stream>>>(t1, w2, g2, b2, m2, v2, t2);

    // Stage 3: shuffle-folded grouped 1x1 GEMM (WMMA f32, 5 acc tiles/wave)
    //          + BN + ReLU + shortcut
    k_conv3<<<dim3(49, 6, 64), dim3(32), 0, stream>>>(t2, w3, g3, b3, m3, v3, x, (float*)d_out);
}